// CoupleClusterLoss_75900662055339
// MI455X (gfx1250) — compile-verified
//
#include <hip/hip_runtime.h>

#define N_SAMPLES 8192
#define DIMS      128
#define N_IDS     1024
#define MARGIN_F  0.3f
#define NCHUNK    8
#define CHUNK     (N_SAMPLES / NCHUNK)   // 1024 samples per block
#define WCHUNK    (CHUNK / 4)            // 256 samples per wave
#define TILES     (WCHUNK / 16)          // 16 j-tiles per wave

typedef __attribute__((ext_vector_type(2))) float v2f;
typedef __attribute__((ext_vector_type(4))) float v4f;
typedef __attribute__((ext_vector_type(8))) float v8f;

__device__ __forceinline__ float pos_inf() { return __int_as_float(0x7f800000); }
__device__ __forceinline__ float neg_inf() { return __int_as_float(0xff800000); }

// ---------------------------------------------------------------- zero init
__global__ void zero_kernel(float* __restrict__ p, int n) {
    int i = blockIdx.x * blockDim.x + threadIdx.x;
    if (i < n) p[i] = 0.0f;
}

// ------------------------------------------------- per-class sum + counts
__global__ void scatter_kernel(const float* __restrict__ inputs,
                               const int*   __restrict__ targets,
                               float* __restrict__ centers,
                               float* __restrict__ counts) {
    int idx = blockIdx.x * blockDim.x + threadIdx.x;
    if (idx >= N_SAMPLES * DIMS) return;
    int i = idx >> 7;            // sample
    int d = idx & (DIMS - 1);    // dim
    int t = targets[i];
    atomicAdd(&centers[t * DIMS + d], inputs[idx]);
    if (d == 0) atomicAdd(&counts[t], 1.0f);
}

// ------------------------------------------------------- x2[j] = |x_j|^2
// one wave per sample, lane loads float4 (coalesced 512B per wave)
__global__ void x2_kernel(const float* __restrict__ inputs,
                          float* __restrict__ x2) {
    int wave = threadIdx.x >> 5;
    int lane = threadIdx.x & 31;
    int s = blockIdx.x * 4 + wave;
    const float4* p = (const float4*)(inputs + (size_t)s * DIMS);
    float4 v = p[lane];
    float ss = v.x * v.x + v.y * v.y + v.z * v.z + v.w * v.w;
    #pragma unroll
    for (int off = 16; off >= 1; off >>= 1) ss += __shfl_xor(ss, off, 32);
    if (lane == 0) x2[s] = ss;
}

// -------------------------------- centers /= count ; c2[t] = |center_t|^2
__global__ void center_norm_kernel(float* __restrict__ centers,
                                   const float* __restrict__ counts,
                                   float* __restrict__ c2) {
    int t = blockIdx.x;
    int d = threadIdx.x;                  // 128 threads
    float cnt = counts[t];
    float v = centers[t * DIMS + d];
    float cv = (cnt > 0.0f) ? (v / cnt) : 0.0f;
    centers[t * DIMS + d] = cv;
    float ss = cv * cv;
    #pragma unroll
    for (int off = 16; off >= 1; off >>= 1) ss += __shfl_xor(ss, off, 32);
    __shared__ float sm[4];
    if ((threadIdx.x & 31) == 0) sm[threadIdx.x >> 5] = ss;
    __syncthreads();
    if (threadIdx.x == 0) c2[t] = sm[0] + sm[1] + sm[2] + sm[3];
}

// ------------- WMMA f32 GEMM (centers x inputs^T) + fused masked min/max
// grid: (N_IDS/16, NCHUNK); block: 128 (4 waves). Wave w covers 256 samples.
//
// K-slot permutation: float4 slot i of lane-half h holds physical K values
// [8i+4h, 8i+4h+4). WMMA 2i consumes (x,y), WMMA 2i+1 consumes (z,w).
// A and B use the identical mapping, so the dot product is exact; this
// enables 16B-aligned global_load_b128 and halves the VMEM op count.
__global__ void dist_kernel(const float* __restrict__ centers,
                            const float* __restrict__ inputs,
                            const float* __restrict__ c2,
                            const float* __restrict__ x2,
                            const int*   __restrict__ targets,
                            float* __restrict__ pmax,
                            float* __restrict__ pmin) {
    const int t0    = blockIdx.x * 16;       // class tile
    const int chunk = blockIdx.y;
    const int wave  = threadIdx.x >> 5;
    const int lane  = threadIdx.x & 31;
    const int laneM = lane & 15;             // A row / B col / C col
    const int koff  = (lane >> 4) << 2;      // K sub-offset: 0 or 4
    const int rbase = (lane >> 4) << 3;      // C rows 0..7 or 8..15

    // Preload full 16x128 A tile (centers): 16 x b128 per lane, 64 VGPRs.
    __attribute__((aligned(16))) v2f aReg[32];
    const float* crow = centers + (size_t)(t0 + laneM) * DIMS;
    #pragma unroll
    for (int i = 0; i < 16; ++i)
        *(v4f*)&aReg[2 * i] = *(const v4f*)(crow + 8 * i + koff);

    float c2r[8];
    #pragma unroll
    for (int v = 0; v < 8; ++v) c2r[v] = c2[t0 + rbase + v];

    float mp[8], mn[8];
    #pragma unroll
    for (int v = 0; v < 8; ++v) { mp[v] = neg_inf(); mn[v] = pos_inf(); }

    const int jbase = chunk * CHUNK + wave * WCHUNK;
    for (int jt = 0; jt < TILES; ++jt) {
        const int j0 = jbase + jt * 16;
        const int j  = j0 + laneM;
        const float x2j = x2[j];
        const int   tj  = targets[j];

        // Stage the whole 16x128 B tile in registers first: the 16 b128
        // loads issue as one clause and overlap with the WMMA chain.
        const float* xrow = inputs + (size_t)j * DIMS;
        __attribute__((aligned(16))) v2f bReg[32];
        #pragma unroll
        for (int i = 0; i < 16; ++i)
            *(v4f*)&bReg[2 * i] = *(const v4f*)(xrow + 8 * i + koff);

        v8f acc = {};
        #pragma unroll
        for (int kk = 0; kk < 32; ++kk) {
            acc = __builtin_amdgcn_wmma_f32_16x16x4_f32(
                /*neg_a=*/false, aReg[kk], /*neg_b=*/false, bReg[kk],
                /*c_mod=*/(short)0, acc, /*reuse_a=*/false, /*reuse_b=*/false);
        }

        #pragma unroll
        for (int v = 0; v < 8; ++v) {
            float dval = c2r[v] + x2j - 2.0f * acc[v];
            if (tj == t0 + rbase + v) mp[v] = fmaxf(mp[v], dval);
            else                      mn[v] = fminf(mn[v], dval);
        }
    }

    // reduce across the 16 lanes of each half (xor masks 1..8 stay in-half)
    #pragma unroll
    for (int off = 8; off >= 1; off >>= 1) {
        #pragma unroll
        for (int v = 0; v < 8; ++v) {
            mp[v] = fmaxf(mp[v], __shfl_xor(mp[v], off, 32));
            mn[v] = fminf(mn[v], __shfl_xor(mn[v], off, 32));
        }
    }

    __shared__ float smax[4][16];
    __shared__ float smin[4][16];
    if (laneM == 0) {
        #pragma unroll
        for (int v = 0; v < 8; ++v) {
            smax[wave][rbase + v] = mp[v];
            smin[wave][rbase + v] = mn[v];
        }
    }
    __syncthreads();
    if (threadIdx.x < 16) {
        const int r = threadIdx.x;
        float M = smax[0][r], m = smin[0][r];
        #pragma unroll
        for (int w = 1; w < 4; ++w) {
            M = fmaxf(M, smax[w][r]);
            m = fminf(m, smin[w][r]);
        }
        pmax[(size_t)(t0 + r) * NCHUNK + chunk] = M;
        pmin[(size_t)(t0 + r) * NCHUNK + chunk] = m;
    }
}

// ------------------------- reduce chunks, count-weighted loss & precision
__global__ void finalize_kernel(const float* __restrict__ pmax,
                                const float* __restrict__ pmin,
                                const float* __restrict__ counts,
                                float* __restrict__ out) {
    const int t = threadIdx.x;               // 1024 threads, one per class
    float mp = neg_inf(), mn = pos_inf();
    #pragma unroll
    for (int c = 0; c < NCHUNK; ++c) {
        mp = fmaxf(mp, pmax[(size_t)t * NCHUNK + c]);
        mn = fminf(mn, pmin[(size_t)t * NCHUNK + c]);
    }
    float cnt = counts[t];
    float lossC = 0.0f, precC = 0.0f;
    if (cnt > 0.0f) {                        // cnt>0 => mp finite (self-match)
        float r = mp - mn + MARGIN_F;        // mn may be +inf -> r=-inf -> 0
        lossC = cnt * (r > 0.0f ? r : 0.0f);
        precC = (mn > mp) ? cnt : 0.0f;
    }
    #pragma unroll
    for (int off = 16; off >= 1; off >>= 1) {
        lossC += __shfl_xor(lossC, off, 32);
        precC += __shfl_xor(precC, off, 32);
    }
    __shared__ float sl[32];
    __shared__ float sp[32];
    const int wave = threadIdx.x >> 5, lane = threadIdx.x & 31;
    if (lane == 0) { sl[wave] = lossC; sp[wave] = precC; }
    __syncthreads();
    if (wave == 0) {
        float L = sl[lane], P = sp[lane];
        #pragma unroll
        for (int off = 16; off >= 1; off >>= 1) {
            L += __shfl_xor(L, off, 32);
            P += __shfl_xor(P, off, 32);
        }
        if (lane == 0) {
            out[0] = L / (float)N_SAMPLES;
            out[1] = P / (float)N_SAMPLES;
        }
    }
}

extern "C" void kernel_launch(void* const* d_in, const int* in_sizes, int n_in,
                              void* d_out, int out_size, void* d_ws, size_t ws_size,
                              hipStream_t stream) {
    const float* inputs  = (const float*)d_in[0];
    const int*   targets = (const int*)d_in[1];
    float*       out     = (float*)d_out;

    // workspace layout (floats): centers | counts | x2 | c2 | pmax | pmin
    float* centers = (float*)d_ws;                 // 1024*128
    float* counts  = centers + N_IDS * DIMS;       // 1024
    float* x2      = counts + N_IDS;               // 8192
    float* c2      = x2 + N_SAMPLES;               // 1024
    float* pmax    = c2 + N_IDS;                   // 1024*NCHUNK
    float* pmin    = pmax + N_IDS * NCHUNK;        // 1024*NCHUNK

    const int zn = N_IDS * DIMS + N_IDS;           // centers + counts contiguous
    zero_kernel<<<(zn + 255) / 256, 256, 0, stream>>>(centers, zn);

    scatter_kernel<<<(N_SAMPLES * DIMS + 255) / 256, 256, 0, stream>>>(
        inputs, targets, centers, counts);

    x2_kernel<<<N_SAMPLES / 4, 128, 0, stream>>>(inputs, x2);

    center_norm_kernel<<<N_IDS, DIMS, 0, stream>>>(centers, counts, c2);

    dist_kernel<<<dim3(N_IDS / 16, NCHUNK), 128, 0, stream>>>(
        centers, inputs, c2, x2, targets, pmax, pmin);

    finalize_kernel<<<1, N_IDS, 0, stream>>>(pmax, pmin, counts, out);
}